// HawkesPointProcess_12412455486044
// MI455X (gfx1250) — compile-verified
//
#include <hip/hip_runtime.h>

typedef __attribute__((ext_vector_type(2))) float v2f;
typedef __attribute__((ext_vector_type(8))) float v8f;

#define TILE 16

__device__ __forceinline__ float softplus_f(float x) {
    // matches jax.nn.softplus: max(x,0) + log1p(exp(-|x|))  (computed once per wave)
    return fmaxf(x, 0.0f) + log1pf(expf(-fabsf(x)));
}

// XOR-butterfly all-lanes sum over wave32 using DS_SWIZZLE_B32 group-of-32 mode:
// offset = (xor_mask << 10) | (or_mask << 5) | and_mask(0x1F)
template <int IMM>
__device__ __forceinline__ float swz_xor_add(float s) {
    int i = __builtin_amdgcn_ds_swizzle(__builtin_bit_cast(int, s), IMM);
    return s + __builtin_bit_cast(float, i);
}
__device__ __forceinline__ float wave_sum32(float s) {
    s = swz_xor_add<0x401F>(s);  // SWAPX16
    s = swz_xor_add<0x201F>(s);  // SWAPX8
    s = swz_xor_add<0x101F>(s);  // SWAPX4
    s = swz_xor_add<0x081F>(s);  // SWAPX2
    s = swz_xor_add<0x041F>(s);  // SWAPX1
    return s;
}

__global__ __launch_bounds__(32) void hawkes_ll_kernel(
    const float* __restrict__ et,    // (N,T) sorted along T
    const float* __restrict__ msk,   // (N,T)
    const float* __restrict__ t0v,   // (N,)
    const float* __restrict__ t1v,   // (N,)
    const float* __restrict__ mu_p,  // (1,)
    const float* __restrict__ al_p,  // (1,)
    const float* __restrict__ be_p,  // (1,)
    float* __restrict__ out,         // (N,)
    int T)
{
    const int lane  = threadIdx.x;   // 0..31, one wave per block
    const int m     = lane & 15;     // matrix row / event-in-tile this lane owns
    const int khalf = lane >> 4;     // which K-half of the A/B fragments

    const float mu_    = softplus_f(mu_p[0]);
    const float alpha_ = softplus_f(al_p[0]);
    const float beta_  = softplus_f(be_p[0]);

    __shared__ float lds_t[TILE];
    __shared__ float lds_v[TILE];
    __shared__ float lds_s[TILE];

    const float* rt = et  + (size_t)blockIdx.x * (size_t)T;
    const float* rm = msk + (size_t)blockIdx.x * (size_t)T;

    // Constant strictly-lower-triangular ones matrix L as WMMA A fragments
    // (ISA 16x4 f32 A layout: lanes 0-15 hold K={k0,k0+1}, lanes 16-31 K={k0+2,k0+3})
    v2f aL[4];
#pragma unroll
    for (int kc = 0; kc < 4; ++kc) {
        const int k0 = kc * 4 + khalf * 2;
        aL[kc].x = (k0     < m) ? 1.0f : 0.0f;
        aL[kc].y = ((k0+1) < m) ? 1.0f : 0.0f;
    }

    float loglik   = 0.0f;  // per-lane partial (lanes 0..15 carry real terms)
    float Q        = 0.0f;  // prefix exp-sum anchored at end_prev (uniform)
    float end_prev = 0.0f;

    const int ntiles = T / TILE;
    for (int tb = 0; tb < ntiles; ++tb) {
        const int base = tb * TILE;
        // all 32 lanes write (two halves write identical values: benign)
        lds_t[m] = rt[base + m];
        __syncthreads();

        const float a0  = lds_t[0];
        const float t15 = lds_t[TILE - 1];
        const float tm  = lds_t[m];

        // v_k = exp(+beta*(t_k - a0)) >= 1, bounded by tile span
        const float vm = __expf(beta_ * (tm - a0));
        lds_v[m] = vm;
        __syncthreads();

        // D = L x V : D[m][*] = sum_{k<m} v_k  (4 chained f32 WMMAs, K=16)
        v8f c = {};
#pragma unroll
        for (int kc = 0; kc < 4; ++kc) {
            const int k0 = kc * 4;
            v2f b;
            b.x = lds_v[k0 + khalf];       // B rows k0 / k0+1
            b.y = lds_v[k0 + 2 + khalf];   // B rows k0+2 / k0+3
            c = __builtin_amdgcn_wmma_f32_16x16x4_f32(
                    false, aL[kc], false, b, (short)0, c, false, false);
        }
        // D layout: lane 0 holds rows 0..7 (col 0), lane 16 rows 8..15
        if (m == 0) {
            const int rbase = khalf * 8;
#pragma unroll
            for (int r = 0; r < 8; ++r) lds_s[rbase + r] = c[r];
        }
        __syncthreads();
        const float Dsum = lds_s[m];

        const float u     = __expf(-beta_ * (tm  - a0));        // <= 1
        const float eprev = __expf(-beta_ * (a0  - end_prev));  // <= 1
        const float e15   = __expf(-beta_ * (t15 - a0));        // <= 1

        // lamb_m = mu + alpha * sum_{j<i} exp(-beta*(t_i - t_j))
        const float lamb = mu_ + alpha_ * (u * (Dsum + Q * eprev));

        const float mkraw = rm[base + m];                  // unconditional load
        const float mk    = (lane < TILE) ? mkraw : 0.0f;  // count each event once
        loglik += mk * __logf(lamb + 1e-8f);

        // advance prefix to this tile's end anchor:
        // Q_new = sum_{j<=tile} exp(-beta*(t15 - t_j)) = e15*(Q*eprev + sum_k v_k)
        const float vv   = (lane < TILE) ? vm : 0.0f;      // register reuse, no LDS reload
        const float Vsum = wave_sum32(vv);
        Q = e15 * (Q * eprev + Vsum);
        end_prev = t15;
        __syncthreads();
    }

    const float LL = wave_sum32(loglik);

    // compensator: (t1-t0)*mu - a/b * (sum_i exp(-b*(t1-t_i))*[m_i] - sum_i m_i)
    const float t1r = t1v[blockIdx.x];
    const float t0r = t0v[blockIdx.x];
    float ksum = 0.0f, msum = 0.0f;
    for (int e = lane; e < T; e += 32) {
        const float mk = rm[e];
        const float tt = rt[e];
        msum += mk;
        ksum += (mk > 0.5f) ? __expf(-beta_ * (t1r - tt)) : 0.0f;
    }
    ksum = wave_sum32(ksum);
    msum = wave_sum32(msum);

    const float comp = (t1r - t0r) * mu_ - (alpha_ / beta_) * (ksum - msum);
    if (lane == 0) out[blockIdx.x] = LL - comp;
}

extern "C" void kernel_launch(void* const* d_in, const int* in_sizes, int n_in,
                              void* d_out, int out_size, void* d_ws, size_t ws_size,
                              hipStream_t stream) {
    const float* et  = (const float*)d_in[0];
    const float* mk  = (const float*)d_in[1];
    const float* t0v = (const float*)d_in[2];
    const float* t1v = (const float*)d_in[3];
    const float* mu  = (const float*)d_in[4];
    const float* al  = (const float*)d_in[5];
    const float* be  = (const float*)d_in[6];
    float* out = (float*)d_out;

    const int N = in_sizes[2];      // 16
    const int T = in_sizes[0] / N;  // 2048 (multiple of 16)

    hawkes_ll_kernel<<<N, 32, 0, stream>>>(et, mk, t0v, t1v, mu, al, be, out, T);
}